// arcAdversarialLoss_57921928954238
// MI455X (gfx1250) — compile-verified
//
#include <hip/hip_runtime.h>
#include <math.h>

// ---------------------------------------------------------------------------
// ArcFace-style adversarial loss, B=1024 rows x C=16384 cols, memory-bound.
// HBM floor: 3 * 64MB reads = 192MB @ 23.3 TB/s ~= 8.2us. Single global pass:
// each thread keeps its 32 "out" logits in VGPRs + pm bits in one uint32, so
// the post-S_neg sweep is register-only. Cross-lane reductions use
// V_WMMA_F32_16X16X4_F32 with an all-ones B matrix (D[m][n] = v_m + v_{m+16}).
// ---------------------------------------------------------------------------

#define C_DIM   16384
#define BLK     512
#define WAVES   (BLK / 32)
#define CHUNKS  8              // float4 chunks per thread
#define PER_THR 32             // elements per thread

// margin/scale constants (float versions of the Python doubles)
__device__ __constant__ float kCosM  = 0.76484218728448843f;  // cos(0.7)
__device__ __constant__ float kSinM  = 0.64421768723769105f;  // sin(0.7)
__device__ __constant__ float kTh    = -0.76484218728448843f; // cos(pi-0.7)
__device__ __constant__ float kMm    = 0.45095238106638373f;  // sin(0.7)*0.7
__device__ __constant__ float kTau   = 0.2f;
__device__ __constant__ float kTauP1 = 1.2f;
__device__ __constant__ float kScale = 16.0f;

typedef float v2f __attribute__((ext_vector_type(2)));
typedef float v8f __attribute__((ext_vector_type(8)));

// Sum v across all 32 lanes of the wave; result valid in every lane.
// A: VGPR0 = v (K0 for lanes 0-15, K2 for lanes 16-31), VGPR1 = 0.
// B: all ones  =>  D[m][n] = v_m + v_{m+16}.
// Per-lane sum of the 8 D VGPRs gives half-sums; one xor-16 shuffle completes.
__device__ __forceinline__ float wave_sum_wmma(float v) {
  v2f a; a[0] = v;    a[1] = 0.0f;
  v2f b; b[0] = 1.0f; b[1] = 1.0f;
  v8f acc = {};
  acc = __builtin_amdgcn_wmma_f32_16x16x4_f32(
      /*neg_a=*/false, a, /*neg_b=*/false, b,
      /*c_mod=*/(short)0, acc, /*reuse_a=*/false, /*reuse_b=*/false);
  float s = acc[0] + acc[1] + acc[2] + acc[3] +
            acc[4] + acc[5] + acc[6] + acc[7];
  s += __shfl_xor(s, 16, 32);
  return s;
}

__global__ __launch_bounds__(BLK)
void arc_loss_rows(const float* __restrict__ inputs,
                   const float* __restrict__ pos_mask,
                   const int*   __restrict__ pos_cam,
                   const int*   __restrict__ targets,
                   float*       __restrict__ row_loss) {
  const int row = blockIdx.x;
  const int tid = threadIdx.x;
  const int tgt = targets[row];

  const float4* __restrict__ in4 = (const float4*)(inputs   + (size_t)row * C_DIM);
  const float4* __restrict__ pm4 = (const float4*)(pos_mask + (size_t)row * C_DIM);
  const int4*   __restrict__ pc4 = (const int4*)  (pos_cam  + (size_t)row * C_DIM);

  float    outv[PER_THR];
  unsigned pmbits = 0u;
  float    s_neg  = 0.0f;
  float    p_sum  = 0.0f;

  // ---------------- pass 1: stream data once, keep logits in VGPRs ---------
#pragma unroll
  for (int i = 0; i < CHUNKS; ++i) {
    const int v  = tid + BLK * i;           // float4 index (lane-coalesced)
    if (i + 1 < CHUNKS) {                   // emits global_prefetch_b8
      __builtin_prefetch(&in4[v + BLK], 0, 0);
      __builtin_prefetch(&pm4[v + BLK], 0, 0);
      __builtin_prefetch(&pc4[v + BLK], 0, 0);
    }
    const float4 x  = in4[v];
    const float4 pm = pm4[v];
    const int4   pc = pc4[v];
    const int    j0 = 4 * v;

#define ARC_ELEM(K, XV, PMV, PCV)                                          \
    {                                                                      \
      const float cc = fminf(fmaxf((XV), -1.0f), 1.0f);                    \
      float val;                                                           \
      if (j0 + (K) == tgt) {                                               \
        const float sn  = sqrtf(fmaxf(1.0f - cc * cc, 0.0f));              \
        const float phi = cc * kCosM - sn * kSinM;                         \
        val = (cc - kTh > 0.0f) ? phi : (cc - kMm);                        \
      } else {                                                             \
        val = ((PCV) > 0) ? fmaf(kTauP1, cc, kTau) : cc;                   \
      }                                                                    \
      const float o = kScale * val;                                        \
      outv[i * 4 + (K)] = o;                                               \
      s_neg = fmaf(1.0f - (PMV), __expf(o), s_neg);                        \
      p_sum += (PMV);                                                      \
      pmbits |= ((PMV) > 0.5f ? 1u : 0u) << (i * 4 + (K));                 \
    }
    ARC_ELEM(0, x.x, pm.x, pc.x)
    ARC_ELEM(1, x.y, pm.y, pc.y)
    ARC_ELEM(2, x.z, pm.z, pc.z)
    ARC_ELEM(3, x.w, pm.w, pc.w)
#undef ARC_ELEM
  }

  // ---------------- block reduction of S_neg and P (WMMA + LDS) ------------
  __shared__ float l_sneg[WAVES];
  __shared__ float l_p[WAVES];
  __shared__ float l_loss[WAVES];
  __shared__ float tot_sneg, tot_p;

  const int wave = tid >> 5;
  const int lane = tid & 31;

  const float wsn = wave_sum_wmma(s_neg);
  const float wps = wave_sum_wmma(p_sum);
  if (lane == 0) { l_sneg[wave] = wsn; l_p[wave] = wps; }
  __syncthreads();
  if (tid == 0) {
    float a = 0.0f, b = 0.0f;
    for (int w = 0; w < WAVES; ++w) { a += l_sneg[w]; b += l_p[w]; }
    tot_sneg = a; tot_p = b;
  }
  __syncthreads();
  const float S = tot_sneg;
  const float P = tot_p;

  // ---------------- pass 2: register-only sweep ----------------------------
  float loss_loc = 0.0f;
#pragma unroll
  for (int k = 0; k < PER_THR; ++k) {
    if ((pmbits >> k) & 1u) {
      const float o = outv[k];
      loss_loc += __logf(S + __expf(o)) - o;
    }
  }

  const float wl = wave_sum_wmma(loss_loc);
  if (lane == 0) l_loss[wave] = wl;
  __syncthreads();
  if (tid == 0) {
    float a = 0.0f;
    for (int w = 0; w < WAVES; ++w) a += l_loss[w];
    row_loss[row] = a / P;
  }
}

// Deterministic 1024 -> 1 reduction: one wave, fixed association, WMMA sum.
__global__ __launch_bounds__(32)
void arc_loss_final(const float* __restrict__ row_loss,
                    float*       __restrict__ out, int rows) {
  float s = 0.0f;
  for (int i = threadIdx.x; i < rows; i += 32) s += row_loss[i];
  const float t = wave_sum_wmma(s);
  if (threadIdx.x == 0) out[0] = t / (float)rows;
}

extern "C" void kernel_launch(void* const* d_in, const int* in_sizes, int n_in,
                              void* d_out, int out_size, void* d_ws, size_t ws_size,
                              hipStream_t stream) {
  const float* inputs  = (const float*)d_in[0];
  const float* pmask   = (const float*)d_in[1];
  const int*   pcam    = (const int*)d_in[2];
  const int*   tgts    = (const int*)d_in[3];
  // d_in[4] (pos_accu) is unused by the reference.

  const int rows = in_sizes[3];          // B = 1024
  float* row_loss = (float*)d_ws;        // rows * 4 bytes of scratch

  arc_loss_rows<<<rows, BLK, 0, stream>>>(inputs, pmask, pcam, tgts, row_loss);
  arc_loss_final<<<1, 32, 0, stream>>>(row_loss, (float*)d_out, rows);
}